// Mamba3IntraBandBlock_69475390980339
// MI455X (gfx1250) — compile-verified
//
#include <hip/hip_runtime.h>
#include <hip/hip_bf16.h>
#include <math.h>

// ---- problem constants ----
#define BBATCH 2
#define TT 256
#define KK 8
#define DD 256
#define HH 4
#define PP 64
#define SS 128
#define NN (BBATCH * KK)      // 16
#define NT (NN * TT)          // 4096
#define DI 256                // d_inner
#define CONV_DIM 512
#define PROJ_DIM 772
#define PROJ_PAD 832          // 52*16 -> unguarded 64-wide tiles

typedef float v2f __attribute__((ext_vector_type(2)));
typedef float v8f __attribute__((ext_vector_type(8)));
typedef unsigned int v4u __attribute__((ext_vector_type(4)));
typedef int v4i __attribute__((ext_vector_type(4)));
typedef int v8i __attribute__((ext_vector_type(8)));

#define WMMA4(a, b, c)                                                         \
  __builtin_amdgcn_wmma_f32_16x16x4_f32(false, (a), false, (b), (short)0, (c), \
                                        false, false)

#if __has_builtin(__builtin_amdgcn_tensor_load_to_lds) && \
    __has_builtin(__builtin_amdgcn_s_wait_tensorcnt)
#define HAVE_TDM 1
#else
#define HAVE_TDM 0
#endif

__device__ __forceinline__ float silu_f(float x) { return x / (1.f + __expf(-x)); }

// ---------------------------------------------------------------------------
// K1: LayerNorm over D + transpose (b,t,k,d) -> u[(b*K+k)*T + t, d]
// ---------------------------------------------------------------------------
__global__ void ln_kernel(const float* __restrict__ z, const float* __restrict__ gamma,
                          const float* __restrict__ beta, float* __restrict__ u) {
  int wave = threadIdx.x >> 5;
  int lane = threadIdx.x & 31;
  int r = blockIdx.x * 8 + wave;
  const float* zr = z + (size_t)r * DD;
  float v[8];
  float s = 0.f, ss = 0.f;
#pragma unroll
  for (int i = 0; i < 8; ++i) {
    v[i] = zr[lane + 32 * i];
    s += v[i];
    ss += v[i] * v[i];
  }
#pragma unroll
  for (int o = 16; o > 0; o >>= 1) {
    s  += __shfl_xor(s, o, 32);
    ss += __shfl_xor(ss, o, 32);
  }
  float mu  = s * (1.f / DD);
  float var = ss * (1.f / DD) - mu * mu;
  float inv = rsqrtf(var + 1e-5f);
  int b = r >> 11, rem = r & 2047, t = rem >> 3, k = rem & 7;
  float* ur = u + ((size_t)((b * KK + k) * TT + t)) * DD;
#pragma unroll
  for (int i = 0; i < 8; ++i) {
    int d = lane + 32 * i;
    ur[d] = (v[i] - mu) * inv * gamma[d] + beta[d];
  }
}

// ---------------------------------------------------------------------------
// K2: pad W_in [256x772] -> Wpad [256x832] (zeros beyond 772), b_in -> bpad
// ---------------------------------------------------------------------------
__global__ void padw_kernel(const float* __restrict__ W, const float* __restrict__ bias,
                            float* __restrict__ Wpad, float* __restrict__ bpad) {
  int idx = blockIdx.x * blockDim.x + threadIdx.x;
  if (idx >= (DD + 1) * PROJ_PAD) return;
  int r = idx / PROJ_PAD;
  int c = idx - r * PROJ_PAD;
  float v;
  if (r < DD) {
    v = (c < PROJ_DIM) ? W[(size_t)r * PROJ_DIM + c] : 0.f;
    Wpad[idx] = v;
  } else {
    bpad[c] = (c < PROJ_DIM) ? bias[c] : 0.f;
  }
}

// ---------------------------------------------------------------------------
// K3: proj = u[4096x256] @ Wpad[256x832] + bpad    (f32 WMMA 16x16x4)
// block = 4 waves -> 64x64 tile; B panel (64x64) staged in LDS, via the
// Tensor Data Mover when available (tensor_load_to_lds + s_wait_tensorcnt).
// ---------------------------------------------------------------------------
__global__ void __launch_bounds__(128) proj_gemm(const float* __restrict__ u,
                                                 const float* __restrict__ Wpad,
                                                 const float* __restrict__ bpad,
                                                 float* __restrict__ proj) {
  __shared__ float ldsW[64 * 64];   // 16 KB
  int wave = threadIdx.x >> 5;
  int lane = threadIdx.x & 31;
  int l15 = lane & 15, hi = lane >> 4;
  int tm = blockIdx.x * 4 + wave;            // M tile 0..255
  int colbase = blockIdx.y * 64;             // 0..768
  int row = tm * 16 + l15;
  v8f c0 = {}, c1 = {}, c2 = {}, c3 = {};
  for (int kb = 0; kb < DD; kb += 64) {
    __syncthreads();                         // previous panel consumed
#if HAVE_TDM
    if (wave == 0) {
      unsigned long long ga =
          (unsigned long long)(const void*)(Wpad + (size_t)kb * PROJ_PAD + colbase);
      unsigned ldsaddr = (unsigned)(unsigned long long)(void*)ldsW;
      v4u g0;
      g0[0] = 1u;                                        // count=1, user mode
      g0[1] = ldsaddr;                                   // lds_addr
      g0[2] = (unsigned)(ga & 0xFFFFFFFFu);              // global_addr[31:0]
      g0[3] = (unsigned)((ga >> 32) & 0x1FFFFFFu) | (2u << 30);  // [56:32] | type=2
      v8i g1;
      g1[0] = (int)(2u << 16);                           // data_size = 4B
      g1[1] = (int)((unsigned)PROJ_PAD << 16);           // tensor_dim0[15:0]
      g1[2] = (int)(((unsigned)PROJ_PAD >> 16) | (256u << 16));  // dim0_hi | dim1_lo
      g1[3] = (int)(64u << 16);                          // dim1_hi=0 | tile_dim0=64
      g1[4] = (int)64u;                                  // tile_dim1=64, tile_dim2=0
      g1[5] = (int)PROJ_PAD;                             // tensor_dim0_stride lo32
      g1[6] = 0;
      g1[7] = 0;
      v4i zz = {0, 0, 0, 0};
      v8i z8 = {0, 0, 0, 0, 0, 0, 0, 0};
      __builtin_amdgcn_tensor_load_to_lds(g0, g1, zz, zz, z8, 0);
      __builtin_amdgcn_s_wait_tensorcnt((short)0);
    }
#else
    for (int i = threadIdx.x; i < 64 * 64; i += 128) {
      int rr = i >> 6, cc = i & 63;
      ldsW[i] = Wpad[(size_t)(kb + rr) * PROJ_PAD + colbase + cc];
    }
#endif
    __syncthreads();
    const float* ur = u + (size_t)row * DD + kb;
    for (int k = 0; k < 64; k += 4) {
      int ka = k + hi * 2;
      v2f a, b;
      a.x = ur[ka];
      a.y = ur[ka + 1];
      const float* w0 = ldsW + ka * 64;
      const float* w1 = w0 + 64;
      b.x = w0[l15];      b.y = w1[l15];      c0 = WMMA4(a, b, c0);
      b.x = w0[16 + l15]; b.y = w1[16 + l15]; c1 = WMMA4(a, b, c1);
      b.x = w0[32 + l15]; b.y = w1[32 + l15]; c2 = WMMA4(a, b, c2);
      b.x = w0[48 + l15]; b.y = w1[48 + l15]; c3 = WMMA4(a, b, c3);
    }
  }
  int rbase = tm * 16 + hi * 8;
#pragma unroll
  for (int v = 0; v < 8; ++v) {
    float* pr = proj + (size_t)(rbase + v) * PROJ_PAD + colbase;
    pr[l15]      = c0[v] + bpad[colbase + l15];
    pr[16 + l15] = c1[v] + bpad[colbase + 16 + l15];
    pr[32 + l15] = c2[v] + bpad[colbase + 32 + l15];
    pr[48 + l15] = c3[v] + bpad[colbase + 48 + l15];
  }
}

// ---------------------------------------------------------------------------
// K4: depthwise causal conv (K=4) over T + bias + SiLU  -> xc[(n*T+t)*512 + c]
// ---------------------------------------------------------------------------
__global__ void conv_silu(const float* __restrict__ proj, const float* __restrict__ cw,
                          const float* __restrict__ cb, float* __restrict__ xc) {
  int idx = blockIdx.x * blockDim.x + threadIdx.x;
  int c = idx & (CONV_DIM - 1);
  int ntt = idx >> 9;
  int t = ntt & (TT - 1);
  int nbase = ntt - t;
  float acc = cb[c];
#pragma unroll
  for (int j = 0; j < 4; ++j) {
    int tt = t - 3 + j;
    if (tt >= 0)
      acc += proj[(size_t)(nbase + tt) * PROJ_PAD + DI + c] * cw[c * 4 + j];
  }
  xc[idx] = silu_f(acc);
}

// ---------------------------------------------------------------------------
// K5: dt = softplus(dt_raw + bias); inclusive cumsums L[t]=sum dt*exp(A),
// Phi[t]=sum dt*theta per (n,h)  (Hillis-Steele in LDS, one block per (n,h))
// ---------------------------------------------------------------------------
__global__ void dtcum_kernel(const float* __restrict__ proj, const float* __restrict__ dt_bias,
                             const float* __restrict__ A_log, const float* __restrict__ theta,
                             float* __restrict__ dtv, float* __restrict__ Lc,
                             float* __restrict__ Pc) {
  int nh = blockIdx.x;          // n*H + h
  int n = nh >> 2, h = nh & 3;
  int t = threadIdx.x;          // 0..255
  __shared__ float sl[TT];
  __shared__ float sp[TT];
  size_t ntt = (size_t)n * TT + t;
  float x = proj[ntt * PROJ_PAD + (DI + CONV_DIM) + h] + dt_bias[h];
  float spv = (x > 20.f) ? x : log1pf(__expf(x));
  dtv[ntt * HH + h] = spv;
  sl[t] = spv * __expf(A_log[h]);
  sp[t] = spv * theta[h];
  __syncthreads();
  for (int o = 1; o < TT; o <<= 1) {
    float al = (t >= o) ? sl[t - o] : 0.f;
    float ap = (t >= o) ? sp[t - o] : 0.f;
    __syncthreads();
    sl[t] += al;
    sp[t] += ap;
    __syncthreads();
  }
  Lc[(size_t)nh * TT + t] = sl[t];
  Pc[(size_t)nh * TT + t] = sp[t];
}

// ---------------------------------------------------------------------------
// K6: per sequence n:  S[t,tau] = sum_s C[t,s] * B[tau,s]   (f32 WMMA)
// only causal blocks (tau-block <= t-block); 64x64 per block, 16x64 per wave
// ---------------------------------------------------------------------------
__global__ void __launch_bounds__(128) sgemm(const float* __restrict__ xc,
                                             float* __restrict__ S) {
  int bt = blockIdx.x, btau = blockIdx.y;
  if (btau > bt) return;                       // strictly-upper blocks unused
  int wave = threadIdx.x >> 5;
  int lane = threadIdx.x & 31;
  int l15 = lane & 15, hi = lane >> 4;
  int n = blockIdx.z;
  int tm = bt * 4 + wave;
  int trow = tm * 16 + l15;
  int cb = btau * 64;
  const float* Ct = xc + ((size_t)n * TT + trow) * CONV_DIM + DI + SS;
  const float* B0 = xc + ((size_t)n * TT + cb + l15) * CONV_DIM + DI;
  const float* B1 = B0 + 16 * CONV_DIM;
  const float* B2 = B0 + 32 * CONV_DIM;
  const float* B3 = B0 + 48 * CONV_DIM;
  v8f c0 = {}, c1 = {}, c2 = {}, c3 = {};
  for (int k = 0; k < SS; k += 4) {
    int ka = k + hi * 2;
    v2f a, b;
    a.x = Ct[ka];
    a.y = Ct[ka + 1];
    b.x = B0[ka]; b.y = B0[ka + 1]; c0 = WMMA4(a, b, c0);
    b.x = B1[ka]; b.y = B1[ka + 1]; c1 = WMMA4(a, b, c1);
    b.x = B2[ka]; b.y = B2[ka + 1]; c2 = WMMA4(a, b, c2);
    b.x = B3[ka]; b.y = B3[ka + 1]; c3 = WMMA4(a, b, c3);
  }
  float* Sn = S + (size_t)n * TT * TT;
  int rbase = tm * 16 + hi * 8;
#pragma unroll
  for (int v = 0; v < 8; ++v) {
    float* sr = Sn + (size_t)(rbase + v) * TT + cb;
    sr[l15]      = c0[v];
    sr[16 + l15] = c1[v];
    sr[32 + l15] = c2[v];
    sr[48 + l15] = c3[v];
  }
}

// ---------------------------------------------------------------------------
// K7: per (n,h):  y[t,p] = sum_{tau<=t} exp(L_tau-L_t)*cos(Phi_t-Phi_tau)
//                          * S[t,tau] * dt_tau * x[tau,h,p]
// masked causal GEMM; decay/phase folded into A-fragment load; k-loop
// truncated at the causal boundary of the wave's 16-row tile.
// ---------------------------------------------------------------------------
__global__ void __launch_bounds__(128) ygemm(const float* __restrict__ S,
                                             const float* __restrict__ Lc,
                                             const float* __restrict__ Pc,
                                             const float* __restrict__ dtv,
                                             const float* __restrict__ xc,
                                             float* __restrict__ y) {
  int wave = threadIdx.x >> 5;
  int lane = threadIdx.x & 31;
  int l15 = lane & 15, hi = lane >> 4;
  int nh = blockIdx.z;          // n*H + h
  int n = nh >> 2, h = nh & 3;
  int tm = blockIdx.x * 4 + wave;   // 0..15
  int trow = tm * 16 + l15;
  const float* Lh = Lc + (size_t)nh * TT;
  const float* Ph = Pc + (size_t)nh * TT;
  float Lt = Lh[trow];
  float Pt = Ph[trow];
  const float* Srow = S + ((size_t)n * TT + trow) * TT;
  v8f c0 = {}, c1 = {}, c2 = {}, c3 = {};
  int kmax = (tm + 1) * 16;
  for (int k = 0; k < kmax; k += 4) {
    int ka = k + hi * 2;
    v2f a, b;
    int t0 = ka, t1 = ka + 1;
    float w0 = __expf(Lh[t0] - Lt) * __cosf(Pt - Ph[t0]) * Srow[t0];
    float w1 = __expf(Lh[t1] - Lt) * __cosf(Pt - Ph[t1]) * Srow[t1];
    a.x = (t0 <= trow) ? w0 : 0.f;
    a.y = (t1 <= trow) ? w1 : 0.f;
    float d0 = dtv[((size_t)n * TT + t0) * HH + h];
    float d1 = dtv[((size_t)n * TT + t1) * HH + h];
    const float* x0 = xc + ((size_t)n * TT + t0) * CONV_DIM + h * PP;
    const float* x1 = x0 + CONV_DIM;
    b.x = d0 * x0[l15];      b.y = d1 * x1[l15];      c0 = WMMA4(a, b, c0);
    b.x = d0 * x0[16 + l15]; b.y = d1 * x1[16 + l15]; c1 = WMMA4(a, b, c1);
    b.x = d0 * x0[32 + l15]; b.y = d1 * x1[32 + l15]; c2 = WMMA4(a, b, c2);
    b.x = d0 * x0[48 + l15]; b.y = d1 * x1[48 + l15]; c3 = WMMA4(a, b, c3);
  }
  int rbase = tm * 16 + hi * 8;
#pragma unroll
  for (int v = 0; v < 8; ++v) {
    float* yr = y + (size_t)(n * TT + rbase + v) * DI + h * PP;
    yr[l15]      = c0[v];
    yr[16 + l15] = c1[v];
    yr[32 + l15] = c2[v];
    yr[48 + l15] = c3[v];
  }
}

// ---------------------------------------------------------------------------
// K8: y += D_skip*x ; y += (U V^T) @_h y   (H=4 mix), in place
// ---------------------------------------------------------------------------
__global__ void mix_skip(const float* __restrict__ xc, const float* __restrict__ U,
                         const float* __restrict__ V, const float* __restrict__ Dsk,
                         float* __restrict__ y) {
  int idx = blockIdx.x * blockDim.x + threadIdx.x;
  int p = idx & (PP - 1);
  int ntt = idx >> 6;
  float yd[HH];
#pragma unroll
  for (int h = 0; h < HH; ++h) {
    float xv = xc[(size_t)ntt * CONV_DIM + h * PP + p];
    yd[h] = y[(size_t)ntt * DI + h * PP + p] + Dsk[h] * xv;
  }
#pragma unroll
  for (int g = 0; g < HH; ++g) {
    float acc = yd[g];
#pragma unroll
    for (int h = 0; h < HH; ++h) {
      float m = 0.f;
#pragma unroll
      for (int r = 0; r < 4; ++r) m = fmaf(U[g * 4 + r], V[h * 4 + r], m);
      acc = fmaf(m, yd[h], acc);
    }
    y[(size_t)ntt * DI + g * PP + p] = acc;
  }
}

// ---------------------------------------------------------------------------
// K9: out = z + (y * silu(zgate)) @ W_out  (f32 WMMA, 16x64 per wave,
// fused residual + transpose back to (b,t,k,d))
// ---------------------------------------------------------------------------
__global__ void __launch_bounds__(128) out_gemm(const float* __restrict__ y,
                                                const float* __restrict__ proj,
                                                const float* __restrict__ Wo,
                                                const float* __restrict__ z,
                                                float* __restrict__ out) {
  int wave = threadIdx.x >> 5;
  int lane = threadIdx.x & 31;
  int l15 = lane & 15, hi = lane >> 4;
  int tm = blockIdx.x * 4 + wave;           // 0..255
  int colbase = blockIdx.y * 64;            // 0..192
  int row = tm * 16 + l15;
  const float* yr = y + (size_t)row * DI;
  const float* gr = proj + (size_t)row * PROJ_PAD;   // zgate cols 0..255
  v8f c0 = {}, c1 = {}, c2 = {}, c3 = {};
  for (int k = 0; k < DI; k += 4) {
    int ka = k + hi * 2;
    v2f a, b;
    a.x = yr[ka] * silu_f(gr[ka]);
    a.y = yr[ka + 1] * silu_f(gr[ka + 1]);
    const float* w0 = Wo + (size_t)ka * DD + colbase;
    const float* w1 = w0 + DD;
    b.x = w0[l15];      b.y = w1[l15];      c0 = WMMA4(a, b, c0);
    b.x = w0[16 + l15]; b.y = w1[16 + l15]; c1 = WMMA4(a, b, c1);
    b.x = w0[32 + l15]; b.y = w1[32 + l15]; c2 = WMMA4(a, b, c2);
    b.x = w0[48 + l15]; b.y = w1[48 + l15]; c3 = WMMA4(a, b, c3);
  }
  int rbase = tm * 16 + hi * 8;
#pragma unroll
  for (int v = 0; v < 8; ++v) {
    int r = rbase + v;                      // r = n*T + t
    int nn = r >> 8;
    int t = r & (TT - 1);
    int b = nn >> 3;
    int kk = nn & 7;
    size_t obase = ((size_t)((b * TT + t) * KK + kk)) * DD + colbase;
    out[obase + l15]      = z[obase + l15]      + c0[v];
    out[obase + 16 + l15] = z[obase + 16 + l15] + c1[v];
    out[obase + 32 + l15] = z[obase + 32 + l15] + c2[v];
    out[obase + 48 + l15] = z[obase + 48 + l15] + c3[v];
  }
}

// ---------------------------------------------------------------------------
extern "C" void kernel_launch(void* const* d_in, const int* in_sizes, int n_in,
                              void* d_out, int out_size, void* d_ws, size_t ws_size,
                              hipStream_t stream) {
  const float* z       = (const float*)d_in[0];
  const float* ln_g    = (const float*)d_in[1];
  const float* ln_b    = (const float*)d_in[2];
  const float* W_in    = (const float*)d_in[3];
  const float* b_in    = (const float*)d_in[4];
  const float* conv_w  = (const float*)d_in[5];
  const float* conv_b  = (const float*)d_in[6];
  const float* dt_bias = (const float*)d_in[7];
  const float* A_log   = (const float*)d_in[8];
  const float* theta   = (const float*)d_in[9];
  const float* D_skip  = (const float*)d_in[10];
  const float* U_mimo  = (const float*)d_in[11];
  const float* V_mimo  = (const float*)d_in[12];
  const float* W_out   = (const float*)d_in[13];
  float* out = (float*)d_out;

  float* ws = (float*)d_ws;
  size_t off = 0;
  float* u    = ws + off; off += (size_t)NT * DD;            // 1,048,576 (aliased by y)
  float* proj = ws + off; off += (size_t)NT * PROJ_PAD;      // 3,407,872
  float* xc   = ws + off; off += (size_t)NT * CONV_DIM;      // 2,097,152
  float* dtv  = ws + off; off += (size_t)NN * TT * HH;       // 16,384
  float* Lc   = ws + off; off += (size_t)NN * HH * TT;       // 16,384
  float* Pc   = ws + off; off += (size_t)NN * HH * TT;       // 16,384
  float* Wpad = ws + off; off += (size_t)DD * PROJ_PAD;      // 212,992
  float* bpad = ws + off; off += (size_t)PROJ_PAD;           // 832
  float* Smat = ws + off; off += (size_t)NN * TT * TT;       // 1,048,576
  float* y = u;  // u is dead after proj_gemm

  ln_kernel<<<NT / 8, 256, 0, stream>>>(z, ln_g, ln_b, u);

  padw_kernel<<<((DD + 1) * PROJ_PAD + 255) / 256, 256, 0, stream>>>(
      W_in, b_in, Wpad, bpad);

  proj_gemm<<<dim3(NT / 64, PROJ_PAD / 64), 128, 0, stream>>>(u, Wpad, bpad, proj);

  conv_silu<<<(NT * CONV_DIM) / 256, 256, 0, stream>>>(proj, conv_w, conv_b, xc);

  dtcum_kernel<<<NN * HH, TT, 0, stream>>>(proj, dt_bias, A_log, theta, dtv, Lc, Pc);

  sgemm<<<dim3(TT / 64, TT / 64, NN), 128, 0, stream>>>(xc, Smat);

  ygemm<<<dim3(TT / 64, 1, NN * HH), 128, 0, stream>>>(Smat, Lc, Pc, dtv, xc, y);

  mix_skip<<<(NT * PP) / 256, 256, 0, stream>>>(xc, U_mimo, V_mimo, D_skip, y);

  out_gemm<<<dim3(NT / 64, DD / 64), 128, 0, stream>>>(y, proj, W_out, z, out);
}